// Ecology_86011015070339
// MI455X (gfx1250) — compile-verified
//
#include <hip/hip_runtime.h>
#include <stdint.h>

// ---------------------------------------------------------------------------
// Problem constants (from reference)
// ---------------------------------------------------------------------------
#define NPART   131072
#define TSTEPS  256
#define MAXTAU  25
#define CHUNK   16                 // steps staged per TDM tile
#define NCHUNK  (TSTEPS / CHUNK)
#define BLOCK   128                // particles per workgroup (4 waves)
#define WINSTR  26                 // padded delay-line stride (banks)
#define LOG2PI_F 1.8378770664093453f

// Flat element offsets inside d_out (tuple return order)
#define OFF_TAU  ((size_t)0)
#define OFF_ET   ((size_t)NPART)
#define OFF_EPS  (OFF_ET  + (size_t)NPART * TSTEPS)
#define OFF_ZS   (OFF_EPS + (size_t)NPART * TSTEPS)
#define OFF_XS   (OFF_ZS  + (size_t)NPART * (MAXTAU + TSTEPS))
#define OFF_LP   (OFF_XS  + (size_t)NPART * TSTEPS)
#define OFF_NAT  (OFF_LP  + (size_t)NPART)

typedef unsigned int u32x4 __attribute__((ext_vector_type(4)));
typedef int          i32x8 __attribute__((ext_vector_type(8)));
typedef int          i32x4 __attribute__((ext_vector_type(4)));
typedef float        f32x4 __attribute__((ext_vector_type(4)));

// ---------------------------------------------------------------------------
// Counter-based RNG (deterministic across graph replays)
// ---------------------------------------------------------------------------
__device__ __forceinline__ uint64_t mix64(uint64_t x) {
  x += 0x9E3779B97F4A7C15ull;
  x = (x ^ (x >> 30)) * 0xBF58476D1CE4E5B9ull;
  x = (x ^ (x >> 27)) * 0x94D049BB133111EBull;
  return x ^ (x >> 31);
}
__device__ __forceinline__ float u01a(uint64_t h) {
  return ((float)(uint32_t)(h >> 40) + 0.5f) * (1.0f / 16777216.0f);
}
__device__ __forceinline__ float u01b(uint64_t h) {
  return ((float)(((uint32_t)h) >> 8) + 0.5f) * (1.0f / 16777216.0f);
}
__device__ __forceinline__ float normal1(uint64_t h) {
  float u1 = u01a(h), u2 = u01b(h);
  float r = __fsqrt_rn(-2.0f * __logf(u1));
  float s, c;
  __sincosf(6.28318530717958647f * u2, &s, &c);
  return r * c;
}
// Marsaglia-Tsang gamma(a>=1); returns un-normalized sample g, *logg = ln g
__device__ __forceinline__ float gamma_draw(uint64_t ctr, float d, float c,
                                            float* logg) {
  uint64_t h1 = mix64(ctr);
  uint64_t h2 = mix64(ctr ^ 0xD2B74407B1CE6E93ull);
  float n = normal1(h1);
  float v = __fmaf_rn(c, n, 1.0f);
  bool ok = v > 0.0f;
  v = ok ? v * v * v : 1.0f;
  float u = u01a(h2);
  bool acc = ok && (__logf(u) < 0.5f * n * n + d - d * v + d * __logf(v));
  if (!acc) {  // single deterministic retry, clamped-accept
    float n2 = normal1(mix64(ctr ^ 0xA5A5A5A5DEADBEEFull));
    float v2 = fmaxf(__fmaf_rn(c, n2, 1.0f), 0.05f);
    v = v2 * v2 * v2;
  }
  float g = d * v;
  *logg = __logf(g);
  return g;
}

// ---------------------------------------------------------------------------
// TDM: async DMA of a row-major [tile_y][tile_x] f32 LDS tile to global with
// row stride `row_stride_elems`. Descriptor per CDNA5 ISA ch.8 (D# groups).
// Tracked on TENSORcnt; issued once per wave (EXEC ignored by TDM).
// This toolchain's builtin takes 6 args (extra i32x8 group before cpol).
// ---------------------------------------------------------------------------
__device__ __forceinline__ void tdm_store_tile_2d(const float* lds_tile,
                                                  const float* gptr,
                                                  uint32_t tile_x,
                                                  uint32_t tile_y,
                                                  uint64_t row_stride_elems) {
  // ISA 10.2: LDS aperture truncates flat address to addr[31:0] == LDS byte
  // offset, so the low 32 bits of the generic pointer are the LDS address.
  uint32_t lds_addr = (uint32_t)(uintptr_t)lds_tile;
  uint64_t ga = (uint64_t)(uintptr_t)gptr;

  u32x4 g0;
  g0.x = 1u;                                             // count=1, user desc
  g0.y = lds_addr;                                       // lds_addr
  g0.z = (uint32_t)ga;                                   // global_addr[31:0]
  g0.w = (uint32_t)((ga >> 32) & 0x01FFFFFFu) | (2u << 30); // ga[56:32]|type=2

  const uint32_t td0 = 0x7FFFFFFFu, td1 = 0x7FFFFFFFu;   // huge dims: no OOB drop
  i32x8 g1;
  g1[0] = (int)(2u << 16);                               // data_size = 4 bytes
  g1[1] = (int)((td0 & 0xFFFFu) << 16);                  // tensor_dim0[15:0]
  g1[2] = (int)((td0 >> 16) | ((td1 & 0xFFFFu) << 16));  // td0 hi | td1 lo
  g1[3] = (int)((td1 >> 16) | ((tile_x & 0xFFFFu) << 16)); // td1 hi | tile_dim0
  g1[4] = (int)(tile_y & 0xFFFFu);                       // tile_dim1; tile_dim2=0
  g1[5] = (int)(uint32_t)(row_stride_elems & 0xFFFFFFFFull); // dim0_stride lo
  g1[6] = (int)(uint32_t)((row_stride_elems >> 32) & 0xFFFFull);
  g1[7] = 0;

  i32x4 g2; g2[0] = 1; g2[1] = 1; g2[2] = 0; g2[3] = 0;  // dim2=dim3=1 (benign)
  i32x4 g3; g3[0] = 0; g3[1] = (int)(1u << 16); g3[2] = 0; g3[3] = 0; // dim4=1
  i32x8 g4; g4[0]=0; g4[1]=0; g4[2]=0; g4[3]=0; g4[4]=0; g4[5]=0; g4[6]=0; g4[7]=0;

#if __has_builtin(__builtin_amdgcn_tensor_store_from_lds)
  __builtin_amdgcn_tensor_store_from_lds(g0, g1, g2, g3, g4, 0);
#endif
}

// ---------------------------------------------------------------------------
// Zero-fill the 256 MB `nat` region with NT b128 stores (pure stream).
// ---------------------------------------------------------------------------
__global__ void nat_zero_kernel(float* __restrict__ out) {
  const size_t n4 = (size_t)NPART * 512 / 4;
  f32x4* base = (f32x4*)(out + OFF_NAT);
  f32x4 z = {0.f, 0.f, 0.f, 0.f};
  size_t stride = (size_t)gridDim.x * blockDim.x;
  for (size_t i = (size_t)blockIdx.x * blockDim.x + threadIdx.x; i < n4;
       i += stride)
    __builtin_nontemporal_store(z, base + i);
}

// ---------------------------------------------------------------------------
// Main scan kernel: 1 lane = 1 particle; CHUNK-step tiles double-buffered in
// LDS and streamed out by the Tensor Data Mover, overlapped with compute.
// ---------------------------------------------------------------------------
__global__ __launch_bounds__(BLOCK) void eco_scan_kernel(
    const float* __restrict__ log_P_p, const float* __restrict__ log_N0_p,
    const float* __restrict__ z0, const float* __restrict__ log_s_d_p,
    const float* __restrict__ log_s_p_p, const float* __restrict__ tau_logits,
    const float* __restrict__ log_delta_p,
    const float* __restrict__ log_rstd_p, float* __restrict__ out) {
  __shared__ __align__(16) float s_sig[MAXTAU];
  __shared__ __align__(16) float s_cdf[MAXTAU];
  __shared__ __align__(16) float s_lsm[MAXTAU];
  __shared__ __align__(16) float s_win[BLOCK * WINSTR];
  __shared__ __align__(16) float s_et[2][BLOCK * CHUNK];
  __shared__ __align__(16) float s_ep[2][BLOCK * CHUNK];
  __shared__ __align__(16) float s_zz[2][BLOCK * CHUNK];

  const int lane = threadIdx.x;
  const int p = blockIdx.x * BLOCK + lane;

  if (lane == 0) {  // 25-wide softmax / CDF / sigmoid tables
    float mx = tau_logits[0];
    for (int j = 1; j < MAXTAU; ++j) mx = fmaxf(mx, tau_logits[j]);
    float s = 0.f;
    for (int j = 0; j < MAXTAU; ++j) {
      float e = __expf(tau_logits[j] - mx);
      s_cdf[j] = e; s += e;
    }
    float inv = 1.f / s, ls = __logf(s), run = 0.f;
    for (int j = 0; j < MAXTAU; ++j) {
      run += s_cdf[j] * inv;
      s_lsm[j] = tau_logits[j] - mx - ls;
      s_cdf[j] = run;
    }
    for (int j = 0; j < MAXTAU; ++j) s_sig[j] = 1.f / (1.f + __expf(-z0[j]));
  }
  __syncthreads();

  // Scalar parameters (uniform scalar loads)
  const float P = __expf(log_P_p[0]);
  const float invN0 = __expf(-log_N0_p[0]);
  const float s_dv = __expf(log_s_d_p[0]);
  const float s_pv = __expf(log_s_p_p[0]);
  const float delta = __expf(log_delta_p[0]);
  const float sigma = __expf(log_rstd_p[0]);
  const float lsig = log_rstd_p[0];
  const float a_p = 1.f / (s_pv * s_pv), a_d = 1.f / (s_dv * s_dv);
  const float d_p = a_p - (1.f / 3.f), c_p = rsqrtf(9.f * d_p);
  const float d_d = a_d - (1.f / 3.f), c_d = rsqrtf(9.f * d_d);
  const float C_p = __logf(a_p) - lgammaf(a_p);
  const float C_d = __logf(a_d) - lgammaf(a_d);
  const float inva_p = 1.f / a_p, inva_d = 1.f / a_d;

  // Categorical tau draw
  float u_tau = u01a(mix64(0x7461753131ull ^
                           ((uint64_t)(uint32_t)p * 0x9E3779B97F4A7C15ull)));
  int tau0 = MAXTAU - 1;
  for (int j = 0; j < MAXTAU - 1; ++j)
    if (u_tau <= s_cdf[j]) { tau0 = j; break; }
  const int tau = tau0 + 1;
  float logp = s_lsm[tau0];

  float z = s_sig[MAXTAU - 1];
  int wr = 0;
  int rd = MAXTAU - tau;  // (t - tau) mod 25 at t=0
  float* win = &s_win[lane * WINSTR];

  // zs prefix row (w0 = sigmoid(z0), identical per particle): NT stream
  {
    float* zrow = out + OFF_ZS + (size_t)p * (MAXTAU + TSTEPS);
    for (int j = 0; j < MAXTAU; ++j)
      __builtin_nontemporal_store(s_sig[j], zrow + j);
  }

  for (int c = 0; c < NCHUNK; ++c) {
    const int buf = c & 1;
    if (c >= 2) {
      // TDM done-signals are in-order per wave: <=4 outstanding => the 4
      // descriptors for this buffer (issued at chunk c-2) have retired.
      if (threadIdx.x < 32) __builtin_amdgcn_s_wait_tensorcnt((short)4);
      __syncthreads();
    }

    float etb[CHUNK], epb[CHUNK], zb[CHUNK];
#pragma unroll
    for (int k = 0; k < CHUNK; ++k) {
      const int t = c * CHUNK + k;
      uint64_t base =
          ((uint64_t)(uint32_t)t << 34) ^ (uint64_t)(uint32_t)p;
      float lg_p, lg_d;
      float et_g = gamma_draw(base ^ 0x1111111111111111ull, d_p, c_p, &lg_p);
      float ep_g = gamma_draw(base ^ 0x2222222222222222ull, d_d, c_d, &lg_d);
      float et_t = et_g * inva_p;   // reference divides gamma(a) by a
      float ep_t = ep_g * inva_d;
      // lp = a ln a + (a-1) ln(g/a) - g - lgamma(a) = C + (a-1) ln g - g
      logp += C_p + (a_p - 1.f) * lg_p - et_g;
      logp += C_d + (a_d - 1.f) * lg_d - ep_g;

      float n = normal1(mix64(base ^ 0x3333333333333333ull));
      float ztm = (t < tau) ? s_sig[MAXTAU - tau + t] : win[rd];
      rd = (rd + 1 == MAXTAU) ? 0 : rd + 1;

      float zdet = P * ztm * __expf(-ztm * invN0) * et_t +
                   z * __expf(-delta * ep_t);
      float mu = __logf(zdet);
      float logz = __fmaf_rn(sigma, n, mu);  // ln(znew) exactly
      float znew = __expf(logz);
      logp += -logz - lsig - 0.5f * LOG2PI_F - 0.5f * n * n;

      win[wr] = znew;
      wr = (wr + 1 == MAXTAU) ? 0 : wr + 1;
      z = znew;
      etb[k] = et_t; epb[k] = ep_t; zb[k] = znew;
    }

    // Stage registers -> LDS tile (row-major [particle][step], b128 writes)
#pragma unroll
    for (int q = 0; q < CHUNK / 4; ++q) {
      f32x4 ve = {etb[4 * q], etb[4 * q + 1], etb[4 * q + 2], etb[4 * q + 3]};
      f32x4 vp = {epb[4 * q], epb[4 * q + 1], epb[4 * q + 2], epb[4 * q + 3]};
      f32x4 vz = {zb[4 * q], zb[4 * q + 1], zb[4 * q + 2], zb[4 * q + 3]};
      ((f32x4*)&s_et[buf][lane * CHUNK])[q] = ve;
      ((f32x4*)&s_ep[buf][lane * CHUNK])[q] = vp;
      ((f32x4*)&s_zz[buf][lane * CHUNK])[q] = vz;
    }
    __syncthreads();  // tile visible to TDM

    if (threadIdx.x < 32) {  // wave 0 issues the DMA descriptors
      const size_t p0 = (size_t)blockIdx.x * BLOCK;
      const int t0 = c * CHUNK;
      tdm_store_tile_2d(&s_et[buf][0], out + OFF_ET + p0 * TSTEPS + t0,
                        CHUNK, BLOCK, TSTEPS);
      tdm_store_tile_2d(&s_ep[buf][0], out + OFF_EPS + p0 * TSTEPS + t0,
                        CHUNK, BLOCK, TSTEPS);
      tdm_store_tile_2d(&s_zz[buf][0], out + OFF_XS + p0 * TSTEPS + t0,
                        CHUNK, BLOCK, TSTEPS);
      tdm_store_tile_2d(&s_zz[buf][0],
                        out + OFF_ZS + p0 * (MAXTAU + TSTEPS) + MAXTAU + t0,
                        CHUNK, BLOCK, MAXTAU + TSTEPS);
    }
  }

  __builtin_nontemporal_store(-logp, out + OFF_LP + (size_t)p);
  __builtin_nontemporal_store(tau, (int*)out + p);  // tau as raw int32 bits

  if (threadIdx.x < 32) __builtin_amdgcn_s_wait_tensorcnt((short)0);
}

// ---------------------------------------------------------------------------
extern "C" void kernel_launch(void* const* d_in, const int* in_sizes, int n_in,
                              void* d_out, int out_size, void* d_ws,
                              size_t ws_size, hipStream_t stream) {
  (void)in_sizes; (void)n_in; (void)out_size; (void)d_ws; (void)ws_size;
  // setup_inputs order: I, N, log_P, log_N0, z_0, log_s_d, log_s_p,
  //                     tau_logits, log_delta, log_rand_std
  const float* log_P = (const float*)d_in[2];
  const float* log_N0 = (const float*)d_in[3];
  const float* z0 = (const float*)d_in[4];
  const float* log_s_d = (const float*)d_in[5];
  const float* log_s_p = (const float*)d_in[6];
  const float* tau_logits = (const float*)d_in[7];
  const float* log_delta = (const float*)d_in[8];
  const float* log_rstd = (const float*)d_in[9];
  float* out = (float*)d_out;

  nat_zero_kernel<<<dim3(1024), dim3(256), 0, stream>>>(out);
  eco_scan_kernel<<<dim3(NPART / BLOCK), dim3(BLOCK), 0, stream>>>(
      log_P, log_N0, z0, log_s_d, log_s_p, tau_logits, log_delta, log_rstd,
      out);
}